// WarpBlock_86887188398979
// MI455X (gfx1250) — compile-verified
//
#include <hip/hip_runtime.h>
#include <cstdint>

// ---------------------------------------------------------------------------
// TV-L1 optical flow forward pass (MI455X / gfx1250).
// Memory-bound stencil/pointwise solver; state (~160MB) fits the 192MB L2, so
// the 10 iterations are L2-resident. WMMA is inapplicable (no contraction);
// the CDNA5 win is TDM (tensor_load_to_lds) staging of u-tiles into LDS for
// the dual (p) update, which reads each u element 3x (self, +x, +y).
// ---------------------------------------------------------------------------

static constexpr int W    = 2048;
static constexpr int H    = 2048;
static constexpr int NPIX = W * H;

static constexpr float LT_F   = (float)(0.15 * 0.3);   // lambda*theta
static constexpr float TAUT_F = (float)(0.25 / 0.3);   // tau/theta
static constexpr float THETA_F = 0.3f;
static constexpr float EPS_F  = 1e-12f;

#define TILE_W 64
#define TILE_H 32

typedef __attribute__((ext_vector_type(4))) unsigned int v4u;
typedef __attribute__((ext_vector_type(8))) int          v8i;
typedef __attribute__((ext_vector_type(4))) int          v4i;

#if defined(__HIP_DEVICE_COMPILE__) && __has_builtin(__builtin_amdgcn_tensor_load_to_lds)
#define USE_TDM 1
#endif

// ---------------------------------------------------------------------------
// Prep: centered grads of x2 computed inline at the 4 bilinear corners,
// warp x2/dx/dy by the initial flow, build rho_c, copy u into d_out planes,
// zero the dual field p. Pure per-pixel writes -> one fused kernel.
// ---------------------------------------------------------------------------
__global__ __launch_bounds__(256) void tvl1_prep(
    const float* __restrict__ x1, const float* __restrict__ x2,
    const float* __restrict__ u1in, const float* __restrict__ u2in,
    float* __restrict__ U1, float* __restrict__ U2,
    float* __restrict__ dxw, float* __restrict__ dyw,
    float* __restrict__ rho_c,
    float* __restrict__ p11, float* __restrict__ p12,
    float* __restrict__ p21, float* __restrict__ p22)
{
    int i = blockIdx.x * blockDim.x + threadIdx.x;
    if (i >= NPIX) return;
    int x = i & (W - 1);
    int y = i >> 11;

    float uu = u1in[i], vv = u2in[i];

    // gx = linspace(-1,1,W)[x] + u/((W-1)/2) ; ix = clip(((gx+1)*W-1)/2, 0, W-1)
    const float stepx = 2.0f / (float)(W - 1);
    const float stepy = 2.0f / (float)(H - 1);
    float gx = fmaf((float)x, stepx, -1.0f) + uu / 1023.5f;
    float gy = fmaf((float)y, stepy, -1.0f) + vv / 1023.5f;
    float ix = fminf(fmaxf(((gx + 1.0f) * (float)W - 1.0f) * 0.5f, 0.0f), (float)(W - 1));
    float iy = fminf(fmaxf(((gy + 1.0f) * (float)H - 1.0f) * 0.5f, 0.0f), (float)(H - 1));

    float x0f = floorf(ix), y0f = floorf(iy);
    float wx = ix - x0f,    wy = iy - y0f;
    int x0 = (int)x0f, y0 = (int)y0f;
    int x1i = min(x0 + 1, W - 1);
    int y1i = min(y0 + 1, H - 1);

    // corner sampler: value + inline centered gradient (boundary forms reduce
    // to clamped-index differences: dx[0]=0.5(x[1]-x[0]), dx[W-1]=0.5(x[W-1]-x[W-2]))
    float v[4], dxv[4], dyv[4];
    int cys[2] = { y0, y1i };
    int cxs[2] = { x0, x1i };
#pragma unroll
    for (int cy = 0; cy < 2; ++cy) {
#pragma unroll
        for (int cx = 0; cx < 2; ++cx) {
            int yy = cys[cy], xx = cxs[cx];
            const float* row = x2 + (size_t)yy * W;
            float c  = row[xx];
            float xr = row[min(xx + 1, W - 1)];
            float xl = row[max(xx - 1, 0)];
            float yd = x2[(size_t)min(yy + 1, H - 1) * W + xx];
            float yu = x2[(size_t)max(yy - 1, 0) * W + xx];
            int k = cy * 2 + cx;
            v[k]   = c;
            dxv[k] = 0.5f * (xr - xl);
            dyv[k] = 0.5f * (yd - yu);
        }
    }
    float w00 = (1.0f - wx) * (1.0f - wy);
    float w01 = wx * (1.0f - wy);
    float w10 = (1.0f - wx) * wy;
    float w11 = wx * wy;

    float x2w = v[0] * w00 + v[1] * w01 + v[2] * w10 + v[3] * w11;
    float dxs = dxv[0] * w00 + dxv[1] * w01 + dxv[2] * w10 + dxv[3] * w11;
    float dys = dyv[0] * w00 + dyv[1] * w01 + dyv[2] * w10 + dyv[3] * w11;

    dxw[i]   = dxs;
    dyw[i]   = dys;
    rho_c[i] = x2w - dxs * uu - dys * vv - x1[i];
    U1[i] = uu;
    U2[i] = vv;
    p11[i] = 0.0f; p12[i] = 0.0f; p21[i] = 0.0f; p22[i] = 0.0f;
}

// ---------------------------------------------------------------------------
// Primal update: rho, thresholded data step, + THETA * div(p). In-place safe:
// u_new[i] depends only on u[i]; p read at {i, i-1, i-W}.
// grad = dxw^2 + dyw^2 + EPS recomputed (2 flops beats a 16MB L2 read/iter).
// rho is only stored on the last iteration (the only one the reference returns).
// ---------------------------------------------------------------------------
__global__ __launch_bounds__(256) void tvl1_update_u(
    float* __restrict__ U1, float* __restrict__ U2, float* __restrict__ RHO,
    const float* __restrict__ dxw, const float* __restrict__ dyw,
    const float* __restrict__ rho_c,
    const float* __restrict__ p11, const float* __restrict__ p12,
    const float* __restrict__ p21, const float* __restrict__ p22,
    int write_rho)
{
    int i = blockIdx.x * blockDim.x + threadIdx.x;
    if (i >= NPIX) return;
    int x = i & (W - 1);
    int y = i >> 11;

    float u1v = U1[i], u2v = U2[i];
    float dx = dxw[i], dy = dyw[i];
    float g  = dx * dx + dy * dy + EPS_F;
    float rho = rho_c[i] + dx * u1v + dy * u2v + EPS_F;
    float th = LT_F * g;

    float d1 = 0.0f, d2 = 0.0f;
    if (rho < -th)      { d1 =  LT_F * dx; d2 =  LT_F * dy; }
    else if (rho > th)  { d1 = -LT_F * dx; d2 = -LT_F * dy; }
    else if (g > EPS_F) { float r = -rho / g; d1 = r * dx; d2 = r * dy; }

    // divergence: dx part from p11/p21 (along W), dy part from p12/p22 (along H)
    float a = p11[i], b = p12[i], c = p21[i], d = p22[i];
    float axm = (x > 0) ? p11[i - 1] : 0.0f;
    float cxm = (x > 0) ? p21[i - 1] : 0.0f;
    float bym = (y > 0) ? p12[i - W] : 0.0f;
    float dym = (y > 0) ? p22[i - W] : 0.0f;
    float div1 = ((x == W - 1) ? -axm : (a - axm)) + ((y == H - 1) ? -bym : (b - bym));
    float div2 = ((x == W - 1) ? -cxm : (c - cxm)) + ((y == H - 1) ? -dym : (d - dym));

    U1[i] = d1 + u1v + THETA_F * div1;
    U2[i] = d2 + u2v + THETA_F * div2;
    if (write_rho) RHO[i] = rho;
}

// ---------------------------------------------------------------------------
// TDM: stage a (TILE_H+1) x (TILE_W+1) fp32 tile of `gbase` into LDS at
// byte-offset lds_off with a single tensor_load_to_lds. tensor_dim = remaining
// rows/cols so bottom/right edge tiles read OOB as zeros (halo gated anyway).
// Descriptor packing per CDNA5 ISA section 8.3/8.4. This toolchain's builtin
// takes 6 args: (g0 v4u, g1 v8i, g2 v4i, g3 v4i, v8i, cpol).
// ---------------------------------------------------------------------------
#ifdef USE_TDM
__device__ inline void tdm_load_tile(const float* gbase, unsigned lds_off,
                                     int x0, int y0)
{
    uint64_t ga  = (uint64_t)(uintptr_t)(gbase + (size_t)y0 * W + x0);
    uint32_t td0 = (uint32_t)(W - x0);
    uint32_t td1 = (uint32_t)(H - y0);
    const uint32_t tile0 = TILE_W + 1;
    const uint32_t tile1 = TILE_H + 1;
    const uint64_t s1 = (uint64_t)W * (uint64_t)H;     // tensor_dim1_stride

    v4u g0 = { 1u,                                           // count=1, user D#
               lds_off,                                      // lds_addr (bytes)
               (uint32_t)ga,                                 // global_addr[31:0]
               ((uint32_t)(ga >> 32) & 0x01FFFFFFu) | (2u << 30) }; // [56:32] | type=2

    v8i g1 = { (int)(2u << 16),                              // data_size = 4B
               (int)((td0 & 0xFFFFu) << 16),                 // tensor_dim0[15:0]
               (int)((td0 >> 16) | ((td1 & 0xFFFFu) << 16)), // td0[31:16] | td1[15:0]
               (int)((td1 >> 16) | (tile0 << 16)),           // td1[31:16] | tile_dim0
               (int)tile1,                                   // tile_dim1, tile_dim2=0
               (int)W,                                       // tensor_dim0_stride[31:0]
               0,                                            // stride0[47:32] | stride1[15:0]
               (int)(uint32_t)(s1 >> 16) };                  // stride1[47:16]

    v4i gz4 = { 0, 0, 0, 0 };                 // groups 2/3 unused for 2D tensor
    v8i gz8 = { 0, 0, 0, 0, 0, 0, 0, 0 };     // trailing group unused
    __builtin_amdgcn_tensor_load_to_lds(g0, g1, gz4, gz4, gz8, 0);
}
#endif

// ---------------------------------------------------------------------------
// Dual update: forward grads of new u (needs +x/+y halo -> TDM-staged LDS
// tile), p updated in place per pixel.
// Block = 256 threads (8 wave32) over a 64x32 tile (halo overhead 1.047x,
// 17.2KB of the 320KB WGP LDS); wave 0 issues both TDM loads, waits
// TENSORcnt, then the workgroup barrier releases everyone.
// ---------------------------------------------------------------------------
__global__ __launch_bounds__(256) void tvl1_update_p(
    const float* __restrict__ U1, const float* __restrict__ U2,
    float* __restrict__ p11, float* __restrict__ p12,
    float* __restrict__ p21, float* __restrict__ p22)
{
    __shared__ __align__(16) float t1[TILE_H + 1][TILE_W + 1];
    __shared__ __align__(16) float t2[TILE_H + 1][TILE_W + 1];

    const int x0 = blockIdx.x * TILE_W;
    const int y0 = blockIdx.y * TILE_H;

#ifdef USE_TDM
    if ((threadIdx.x >> 5) == 0) {   // wave 0 only (s_cbranch_execz skips others)
        unsigned off1 = (unsigned)(uintptr_t)&t1[0][0];  // low 32 bits = LDS byte addr
        unsigned off2 = (unsigned)(uintptr_t)&t2[0][0];
        tdm_load_tile(U1, off1, x0, y0);
        tdm_load_tile(U2, off2, x0, y0);
#if __has_builtin(__builtin_amdgcn_s_wait_tensorcnt)
        __builtin_amdgcn_s_wait_tensorcnt(0);
#else
        asm volatile("s_wait_tensorcnt 0" ::: "memory");
#endif
    }
#else
    for (int idx = threadIdx.x; idx < (TILE_H + 1) * (TILE_W + 1); idx += 256) {
        int ty = idx / (TILE_W + 1), tx = idx % (TILE_W + 1);
        int gy = min(y0 + ty, H - 1), gx = min(x0 + tx, W - 1);
        t1[ty][tx] = U1[(size_t)gy * W + gx];
        t2[ty][tx] = U2[(size_t)gy * W + gx];
    }
#endif
    __syncthreads();

    const int tx = threadIdx.x & (TILE_W - 1);
    const int ty0 = threadIdx.x >> 6;           // 256 threads -> 4 tile rows / pass
#pragma unroll
    for (int k = 0; k < TILE_H / 4; ++k) {
        int ty = ty0 + k * 4;
        int gx = x0 + tx;
        int gy = y0 + ty;
        int i  = gy * W + gx;

        float c1 = t1[ty][tx];
        float r1 = (gx < W - 1) ? (t1[ty][tx + 1] - c1) : 0.0f;
        float b1 = (gy < H - 1) ? (t1[ty + 1][tx] - c1) : 0.0f;
        float c2 = t2[ty][tx];
        float r2 = (gx < W - 1) ? (t2[ty][tx + 1] - c2) : 0.0f;
        float b2 = (gy < H - 1) ? (t2[ty + 1][tx] - c2) : 0.0f;

        float n1 = 1.0f + TAUT_F * sqrtf(r1 * r1 + b1 * b1 + EPS_F);
        float n2 = 1.0f + TAUT_F * sqrtf(r2 * r2 + b2 * b2 + EPS_F);

        p11[i] = (p11[i] + TAUT_F * r1) / n1;
        p12[i] = (p12[i] + TAUT_F * b1) / n1;
        p21[i] = (p21[i] + TAUT_F * r2) / n2;
        p22[i] = (p22[i] + TAUT_F * b2) / n2;
    }
}

// ---------------------------------------------------------------------------
extern "C" void kernel_launch(void* const* d_in, const int* in_sizes, int n_in,
                              void* d_out, int out_size, void* d_ws, size_t ws_size,
                              hipStream_t stream)
{
    const float* x1 = (const float*)d_in[0];
    const float* x2 = (const float*)d_in[1];
    const float* u1 = (const float*)d_in[2];
    const float* u2 = (const float*)d_in[3];

    float* out = (float*)d_out;
    float* U1  = out;                 // output plane 0 doubles as working u1
    float* U2  = out + NPIX;          // plane 1 = working u2
    float* RHO = out + 2 * (size_t)NPIX;

    float* ws    = (float*)d_ws;      // 7 planes * 16MB = 112MB scratch
    float* dxw   = ws + 0 * (size_t)NPIX;
    float* dyw   = ws + 1 * (size_t)NPIX;
    float* rho_c = ws + 2 * (size_t)NPIX;
    float* p11   = ws + 3 * (size_t)NPIX;
    float* p12   = ws + 4 * (size_t)NPIX;
    float* p21   = ws + 5 * (size_t)NPIX;
    float* p22   = ws + 6 * (size_t)NPIX;

    dim3 blk(256);
    dim3 grd((NPIX + 255) / 256);
    tvl1_prep<<<grd, blk, 0, stream>>>(x1, x2, u1, u2, U1, U2,
                                       dxw, dyw, rho_c, p11, p12, p21, p22);

    dim3 pgrd(W / TILE_W, H / TILE_H);
    for (int it = 0; it < 10; ++it) {
        tvl1_update_u<<<grd, blk, 0, stream>>>(U1, U2, RHO, dxw, dyw, rho_c,
                                               p11, p12, p21, p22,
                                               (it == 9) ? 1 : 0);
        tvl1_update_p<<<pgrd, blk, 0, stream>>>(U1, U2, p11, p12, p21, p22);
    }
}